// ASH1DSelfAttention_64957085384892
// MI455X (gfx1250) — compile-verified
//
#include <hip/hip_runtime.h>
#include <stdint.h>

#define BB 2
#define TT 1024
#define EE 64
#define HH 8
#define KKk 8
#define VS 80

typedef __attribute__((ext_vector_type(2))) float v2f;
typedef __attribute__((ext_vector_type(8))) float v8f;

// ---------------- JAX Threefry-2x32 (exact) ----------------
__device__ __forceinline__ uint32_t rotl32(uint32_t x, int n) {
  return (x << n) | (x >> (32 - n));
}

__device__ __forceinline__ void tf2x32(uint32_t k0, uint32_t k1,
                                       uint32_t c0, uint32_t c1,
                                       uint32_t& o0, uint32_t& o1) {
  uint32_t ks2 = k0 ^ k1 ^ 0x1BD11BDAu;
  uint32_t x0 = c0 + k0, x1 = c1 + k1;
#define TFR(r) { x0 += x1; x1 = rotl32(x1, r); x1 ^= x0; }
  TFR(13) TFR(15) TFR(26) TFR(6)   x0 += k1;  x1 += ks2 + 1u;
  TFR(17) TFR(29) TFR(16) TFR(24)  x0 += ks2; x1 += k0  + 2u;
  TFR(13) TFR(15) TFR(26) TFR(6)   x0 += k0;  x1 += k1  + 3u;
  TFR(17) TFR(29) TFR(16) TFR(24)  x0 += k1;  x1 += ks2 + 4u;
  TFR(13) TFR(15) TFR(26) TFR(6)   x0 += ks2; x1 += k0  + 5u;
#undef TFR
  o0 = x0; o1 = x1;
}

// random_bits for an array of n=65536 counters (counter-halves scheme)
__device__ __forceinline__ uint32_t jax_bits65536(uint32_t k0, uint32_t k1, uint32_t f) {
  uint32_t o0, o1;
  if (f < 32768u) { tf2x32(k0, k1, f, f + 32768u, o0, o1); return o0; }
  tf2x32(k0, k1, f - 32768u, f, o0, o1); return o1;
}

__device__ __forceinline__ float softplusf(float x) {
  return fmaxf(x, 0.f) + log1pf(expf(-fabsf(x)));
}

// ---------------- Kernel 1: hyper-net + candidate indices + weights ----------------
__global__ void hyper_kernel(const float* __restrict__ x,
                             const float* __restrict__ Wp1, const float* __restrict__ bp1,
                             const float* __restrict__ Wp2, const float* __restrict__ bp2,
                             int* __restrict__ idx_out, float* __restrict__ w_out) {
  __shared__ float xs[EE + 1];
  __shared__ float hid[256];
  __shared__ float par[16];
  __shared__ float means[KKk], sigmas[KKk];
  __shared__ int   fls[KKk];
  __shared__ int   idxs[VS];
  __shared__ float ps[VS * KKk];
  __shared__ float csum[KKk];

  const int bt  = blockIdx.x;
  const int t   = bt & (TT - 1);
  const int tid = threadIdx.x;

  if (tid < EE) xs[tid] = x[(size_t)bt * EE + tid];
  if (tid == EE) xs[EE] = (float)t / (float)TT;
  __syncthreads();

  {
    float acc = bp1[tid];
    const float* wrow = Wp1 + (size_t)tid * (EE + 1);
    #pragma unroll 5
    for (int i = 0; i < EE + 1; ++i) acc += xs[i] * wrow[i];
    hid[tid] = fmaxf(acc, 0.f);
  }
  __syncthreads();

  if (tid < 16) {
    float acc = bp2[tid];
    const float* wrow = Wp2 + (size_t)tid * 256;
    #pragma unroll 4
    for (int i = 0; i < 256; ++i) acc += hid[i] * wrow[i];
    par[tid] = acc;
  }
  __syncthreads();

  if (tid < KKk) {
    float m = (float)t - 3.0f * softplusf(par[tid]);          // MMULT=3
    m = fminf(fmaxf(m, 0.f), (float)(TT - 1));
    means[tid]  = m;
    fls[tid]    = (int)floorf(m);
    sigmas[tid] = (softplusf(par[tid + KKk] + 2.0f) + 0.05f) * (float)TT * 0.1f;
  }
  __syncthreads();

  if (tid < VS) {
    const int k = tid / 10, j = tid % 10;
    const int fl = fls[k];
    // key(42) = (0,42); split -> kg=(g0,g1), kr=(r0,r1) (constant-folds)
    uint32_t g0, g1, r0, r1;
    tf2x32(0u, 42u, 0u, 2u, g0, r0);
    tf2x32(0u, 42u, 1u, 3u, g1, r1);
    int idx;
    if (j == 0)      idx = fl;
    else if (j == 1) idx = min(fl + 1, TT - 1);
    else if (j < 6) { // global samples; span 1024 (pow2) -> bits & 1023 (matches JAX randint)
      uint32_t f = ((uint32_t)bt * KKk + (uint32_t)k) * 4u + (uint32_t)(j - 2);
      idx = (int)(jax_bits65536(g0, g1, f) & (uint32_t)(TT - 1));
    } else {          // regional samples; span 64 -> bits & 63
      uint32_t f = ((uint32_t)bt * KKk + (uint32_t)k) * 4u + (uint32_t)(j - 6);
      int roff = (int)(jax_bits65536(r0, r1, f) & 63u);
      idx = min(max(fl - 32 + roff, 0), TT - 1);
    }
    idxs[tid] = idx;
  }
  __syncthreads();

  if (tid < VS) {
    const int idx = idxs[tid];
    bool masked = (idx > t);                       // causal
    for (int u = 0; u < tid; ++u)                  // keep-first dedup
      if (idxs[u] == idx) { masked = true; break; }
    const float fi = (float)idx;
    #pragma unroll
    for (int kp = 0; kp < KKk; ++kp) {
      float z = (fi - means[kp]) / sigmas[kp];
      ps[tid * KKk + kp] = masked ? 0.f : expf(-0.5f * z * z);
    }
  }
  __syncthreads();

  if (tid < KKk) {
    float s = 0.f;
    for (int v = 0; v < VS; ++v) s += ps[v * KKk + tid];
    csum[tid] = s;
  }
  __syncthreads();

  if (tid < VS) {
    float w = 0.f;
    #pragma unroll
    for (int kp = 0; kp < KKk; ++kp) w += ps[tid * KKk + kp] / csum[kp];
    w_out[(size_t)bt * VS + tid]  = w;
    idx_out[(size_t)bt * VS + tid] = idxs[tid];
  }
}

// ---------------- Kernel 2: q/k/v projections via fp32 WMMA ----------------
// out[m][n] = sum_k X[m][k] * W[n][k];  one wave per 16x16 tile, K=64 -> 16 wmma steps
__global__ void proj_kernel(const float* __restrict__ x,
                            const float* __restrict__ Wq,
                            const float* __restrict__ Wk,
                            const float* __restrict__ Wv,
                            float* __restrict__ qws, float* __restrict__ kws,
                            float* __restrict__ vws) {
  const int tid    = threadIdx.x;
  const int wave   = tid >> 5;
  const int lane   = tid & 31;
  const int lane16 = lane & 15;
  const int khalf  = lane >> 4;

  const int tile = blockIdx.x * 4 + wave;   // 0 .. 3*128*32-1
  const int mat  = tile / (128 * 32);
  const int rem  = tile - mat * (128 * 32);
  const int m0   = (rem >> 5) * 16;
  const int n0   = (rem & 31) * 16;

  const float* W   = (mat == 0) ? Wq : (mat == 1 ? Wk : Wv);
  float*       dst = (mat == 0) ? qws : (mat == 1 ? kws : vws);
  const float  sc  = (mat < 2) ? 0.35355339059327373f : 1.0f;  // 1/E^0.25 for q,k

  // A 16x4 fp32 layout: lanes 0-15 hold K 0/1, lanes 16-31 hold K 2/3 (per lane: float2)
  const float* arow = x + (size_t)(m0 + lane16) * EE + 2 * khalf;
  const float* brow = W + (size_t)(n0 + lane16) * EE + 2 * khalf;   // B[k][n] = W[n][k]

  v8f acc = {};
  #pragma unroll
  for (int kk = 0; kk < EE; kk += 4) {
    v2f a, b;
    a.x = arow[kk]; a.y = arow[kk + 1];
    b.x = brow[kk]; b.y = brow[kk + 1];
    acc = __builtin_amdgcn_wmma_f32_16x16x4_f32(false, a, false, b, (short)0, acc,
                                                false, false);
  }

  // C/D 16x16 f32: VGPR r -> M=r (lanes 0-15) / M=r+8 (lanes 16-31), N = lane%16
  const int n = n0 + lane16;
  const int h = n >> 6, e = n & 63;
  #pragma unroll
  for (int r = 0; r < 8; ++r) {
    const int m = m0 + r + khalf * 8;
    const int b = m >> 10, t = m & (TT - 1);
    dst[(((size_t)b * HH + h) * TT + t) * EE + e] = acc[r] * sc;
  }
}

// ---------------- Kernel 3: gathered sparse attention (VALU, wave32) ----------------
__global__ void attn_kernel(const float* __restrict__ qws,
                            const float* __restrict__ kws,
                            const float* __restrict__ vws,
                            const int* __restrict__ idx_in,
                            const float* __restrict__ w_in,
                            float* __restrict__ attws) {
  __shared__ float qs[HH * EE];
  __shared__ float att[HH * VS];
  __shared__ int   idxs[VS];
  __shared__ float wgt[VS];

  const int bt  = blockIdx.x;
  const int b   = bt >> 10;
  const int t   = bt & (TT - 1);
  const int tid = threadIdx.x;

  for (int i = tid; i < HH * EE; i += 256) {
    const int h = i >> 6, e = i & 63;
    qs[i] = qws[(((size_t)b * HH + h) * TT + t) * EE + e];
  }
  if (tid < VS) {
    idxs[tid] = idx_in[(size_t)bt * VS + tid];
    wgt[tid]  = w_in[(size_t)bt * VS + tid];
  }
  __syncthreads();

  const int h    = tid >> 5;            // wave = head
  const int lane = tid & 31;
  const float* qrow = qs + h * EE;
  const size_t khbase = ((size_t)b * HH + h) * TT;

  float lg[3];
  #pragma unroll
  for (int r = 0; r < 3; ++r) {
    const int v = lane + r * 32;
    float L = -3.4e38f;
    if (v < VS) {
      const float* krow = kws + (khbase + (size_t)idxs[v]) * EE;
      float d = 0.f;
      #pragma unroll 8
      for (int e = 0; e < EE; ++e) d += qrow[e] * krow[e];
      L = wgt[v] * d;                    // masked entries: weight==0 -> logit 0 (as ref)
    }
    lg[r] = L;
  }
  float mx = fmaxf(fmaxf(lg[0], lg[1]), lg[2]);
  #pragma unroll
  for (int off = 16; off > 0; off >>= 1) mx = fmaxf(mx, __shfl_xor(mx, off, 32));
  float ex[3]; float s = 0.f;
  #pragma unroll
  for (int r = 0; r < 3; ++r) {
    const int v = lane + r * 32;
    float e_ = (v < VS) ? expf(lg[r] - mx) : 0.f;
    ex[r] = e_; s += e_;
  }
  #pragma unroll
  for (int off = 16; off > 0; off >>= 1) s += __shfl_xor(s, off, 32);
  const float inv = 1.f / s;
  #pragma unroll
  for (int r = 0; r < 3; ++r) {
    const int v = lane + r * 32;
    if (v < VS) att[h * VS + v] = ex[r] * inv;
  }
  __syncthreads();

  float a0 = 0.f, a1 = 0.f;
  for (int v = 0; v < VS; ++v) {
    const float a = att[h * VS + v];
    const float* vrow = vws + (khbase + (size_t)idxs[v]) * EE;
    a0 += a * vrow[lane];
    a1 += a * vrow[lane + 32];
  }
  attws[(size_t)bt * (HH * EE) + h * EE + lane]      = a0;   // (b,t,h*E+e)
  attws[(size_t)bt * (HH * EE) + h * EE + lane + 32] = a1;
}

// ---------------- Kernel 4: output projection via fp32 WMMA + bias ----------------
__global__ void outproj_kernel(const float* __restrict__ attws,
                               const float* __restrict__ Wu,
                               const float* __restrict__ bu,
                               float* __restrict__ out) {
  const int tid    = threadIdx.x;
  const int wave   = tid >> 5;
  const int lane   = tid & 31;
  const int lane16 = lane & 15;
  const int khalf  = lane >> 4;

  const int tile = blockIdx.x * 4 + wave;  // 0..511
  const int m0   = (tile >> 2) * 16;
  const int n0   = (tile & 3) * 16;

  const float* arow = attws + (size_t)(m0 + lane16) * (HH * EE) + 2 * khalf;
  const float* brow = Wu    + (size_t)(n0 + lane16) * (HH * EE) + 2 * khalf;

  v8f acc = {};
  #pragma unroll 4
  for (int kk = 0; kk < HH * EE; kk += 4) {
    v2f a, b;
    a.x = arow[kk]; a.y = arow[kk + 1];
    b.x = brow[kk]; b.y = brow[kk + 1];
    acc = __builtin_amdgcn_wmma_f32_16x16x4_f32(false, a, false, b, (short)0, acc,
                                                false, false);
  }

  const int n = n0 + lane16;
  const float bias = bu[n];
  #pragma unroll
  for (int r = 0; r < 8; ++r) {
    const int m = m0 + r + khalf * 8;
    out[(size_t)m * EE + n] = acc[r] + bias;
  }
}

// ---------------- Host launcher ----------------
extern "C" void kernel_launch(void* const* d_in, const int* in_sizes, int n_in,
                              void* d_out, int out_size, void* d_ws, size_t ws_size,
                              hipStream_t stream) {
  (void)in_sizes; (void)n_in; (void)out_size; (void)ws_size;
  const float* x   = (const float*)d_in[0];
  const float* Wk  = (const float*)d_in[1];
  const float* Wq  = (const float*)d_in[2];
  const float* Wv  = (const float*)d_in[3];
  const float* Wu  = (const float*)d_in[4];
  const float* bu  = (const float*)d_in[5];
  const float* Wp1 = (const float*)d_in[6];
  const float* bp1 = (const float*)d_in[7];
  const float* Wp2 = (const float*)d_in[8];
  const float* bp2 = (const float*)d_in[9];
  float* out = (float*)d_out;

  float* qws   = (float*)d_ws;
  float* kws   = qws   + (size_t)BB * HH * TT * EE;
  float* vws   = kws   + (size_t)BB * HH * TT * EE;
  float* attws = vws   + (size_t)BB * HH * TT * EE;
  float* w_ws  = attws + (size_t)BB * TT * HH * EE;
  int*   i_ws  = (int*)(w_ws + (size_t)BB * TT * VS);

  hipLaunchKernelGGL(hyper_kernel, dim3(BB * TT), dim3(256), 0, stream,
                     x, Wp1, bp1, Wp2, bp2, i_ws, w_ws);
  hipLaunchKernelGGL(proj_kernel, dim3(3 * 128 * 32 / 4), dim3(128), 0, stream,
                     x, Wq, Wk, Wv, qws, kws, vws);
  hipLaunchKernelGGL(attn_kernel, dim3(BB * TT), dim3(256), 0, stream,
                     qws, kws, vws, i_ws, w_ws, attws);
  hipLaunchKernelGGL(outproj_kernel, dim3(128), dim3(128), 0, stream,
                     attws, Wu, bu, out);
}